// StateScanPlugin_64922725646536
// MI455X (gfx1250) — compile-verified
//
#include <hip/hip_runtime.h>
#include <hip/hip_bf16.h>

#define BD 32      // batch
#define TT 512     // seq len
#define DD 64      // d_model
#define VV 60      // vocab
#define CHUNK 32   // tokens per chunk-product wave
#define NCHUNK (TT / CHUNK)

typedef __attribute__((ext_vector_type(2))) float v2f;
typedef __attribute__((ext_vector_type(8))) float v8f;

// ---------------------------------------------------------------------------
// Stage A: Modified Gram-Schmidt QR of the 60 distinct vocab matrices.
// One block (64 threads) per vocab entry; thread t owns column t.
// Q stored row-major to d_ws.
// ---------------------------------------------------------------------------
__global__ void ssp_qr_kernel(const float* __restrict__ repr, float* __restrict__ Qg) {
    __shared__ float A[DD * DD];   // column-major: A[col*DD + row]
    const int v = blockIdx.x;
    const int t = threadIdx.x;     // 0..63, owns column t

    #pragma unroll 4
    for (int r = 0; r < DD; ++r)
        A[t * DD + r] = repr[v * DD * DD + r * DD + t];
    __syncthreads();

    for (int j = 0; j < DD; ++j) {
        if (t == j) {
            float s = 0.f;
            #pragma unroll 4
            for (int r = 0; r < DD; ++r) { float x = A[j * DD + r]; s += x * x; }
            float inv = rsqrtf(fmaxf(s, 1e-30f));
            #pragma unroll 4
            for (int r = 0; r < DD; ++r) A[j * DD + r] *= inv;
        }
        __syncthreads();
        if (t > j) {
            float dot = 0.f;
            #pragma unroll 4
            for (int r = 0; r < DD; ++r) dot += A[j * DD + r] * A[t * DD + r];
            #pragma unroll 4
            for (int r = 0; r < DD; ++r) A[t * DD + r] -= dot * A[j * DD + r];
        }
        __syncthreads();
    }

    #pragma unroll 4
    for (int r = 0; r < DD; ++r)
        Qg[v * DD * DD + r * DD + t] = A[t * DD + r];
}

// ---------------------------------------------------------------------------
// Single-wave 64x64x64 fp32 matmul: Snext = A @ Scur using V_WMMA_F32_16X16X4_F32.
// A: row-major, global memory (64x64).
// Scur/Snext: column-major in LDS (S[col*64 + row]).
// Fragment layouts per CDNA5 ISA 7.12.2 (32-bit A 16x4, 32-bit C/D 16x16).
// kk loop kept rolled: one iteration already has 8 loads + 16 WMMAs of ILP,
// and full unroll pushed live VGPRs past the budget (spills observed).
// ---------------------------------------------------------------------------
__device__ __forceinline__ void ssp_wave_mm64(const float* __restrict__ Ag,
                                              const float* __restrict__ Scur,
                                              float* __restrict__ Snext) {
    const int lane = threadIdx.x & 31;
    const int half = lane >> 4;    // 0: lanes 0-15, 1: lanes 16-31
    const int lm   = lane & 15;

    v8f acc[4][4];
    #pragma unroll
    for (int i = 0; i < 4; ++i)
        #pragma unroll
        for (int j = 0; j < 4; ++j)
            acc[i][j] = (v8f){0.f, 0.f, 0.f, 0.f, 0.f, 0.f, 0.f, 0.f};

    #pragma unroll 1
    for (int kk = 0; kk < 16; ++kk) {
        const int kbase = kk * 4 + half * 2;   // two consecutive K values per lane
        v2f afrag[4], bfrag[4];
        #pragma unroll
        for (int i = 0; i < 4; ++i)            // A tile (i,kk): row = i*16+lm, k = kbase..kbase+1
            afrag[i] = *(const v2f*)(Ag + (i * 16 + lm) * DD + kbase);
        #pragma unroll
        for (int j = 0; j < 4; ++j)            // B tile (kk,j): col = j*16+lm, k = kbase..kbase+1
            bfrag[j] = *(const v2f*)(Scur + (j * 16 + lm) * DD + kbase);
        #pragma unroll
        for (int i = 0; i < 4; ++i)
            #pragma unroll
            for (int j = 0; j < 4; ++j)
                acc[i][j] = __builtin_amdgcn_wmma_f32_16x16x4_f32(
                    /*neg_a=*/false, afrag[i],
                    /*neg_b=*/false, bfrag[j],
                    /*c_mod=*/(short)0, acc[i][j],
                    /*reuse_a=*/false, /*reuse_b=*/false);
    }

    // D tile (i,j): vgpr r holds element [i*16 + 8*half + r][j*16 + lm]
    // -> column-major store: 8 consecutive floats per lane.
    #pragma unroll
    for (int i = 0; i < 4; ++i)
        #pragma unroll
        for (int j = 0; j < 4; ++j)
            *(v8f*)(Snext + (j * 16 + lm) * DD + i * 16 + 8 * half) = acc[i][j];
}

// ---------------------------------------------------------------------------
// Stage B: chunk products. grid = (NCHUNK, B), block = 1 wave (32 threads).
// P[b][c] = Q[id[b, c*CHUNK + 31]] @ ... @ Q[id[b, c*CHUNK]]  (row-major out).
// ---------------------------------------------------------------------------
__global__ void __launch_bounds__(32, 1)
ssp_chunk_kernel(const int* __restrict__ ids,
                 const float* __restrict__ Qg,
                 float* __restrict__ Pg) {
    __shared__ float Sbuf[2][DD * DD];
    const int c = blockIdx.x, b = blockIdx.y;
    const int base = b * TT + c * CHUNK;
    const int lane = threadIdx.x;

    // S = Q[ids[base]]  (transpose row-major global -> column-major LDS)
    const float* Q0 = Qg + ids[base] * (DD * DD);
    for (int col = lane; col < DD; col += 32)
        #pragma unroll 4
        for (int r = 0; r < DD; ++r)
            Sbuf[0][col * DD + r] = Q0[r * DD + col];
    __syncthreads();

    int cur = 0;
    #pragma unroll 1
    for (int t = 1; t < CHUNK; ++t) {
        const float* A = Qg + ids[base + t] * (DD * DD);
        ssp_wave_mm64(A, Sbuf[cur], Sbuf[cur ^ 1]);
        __syncthreads();
        cur ^= 1;
    }

    float* out = Pg + (b * NCHUNK + c) * (DD * DD);
    for (int col = lane; col < DD; col += 32)
        #pragma unroll 4
        for (int r = 0; r < DD; ++r)
            out[r * DD + col] = Sbuf[cur][col * DD + r];
}

// ---------------------------------------------------------------------------
// Stage C: combine the NCHUNK partials per batch. grid = B, block = 1 wave.
// S_b = P[b][NCHUNK-1] @ ... @ P[b][0]  (row-major out).
// ---------------------------------------------------------------------------
__global__ void __launch_bounds__(32, 1)
ssp_combine_kernel(const float* __restrict__ Pg,
                   float* __restrict__ Sout) {
    __shared__ float Sbuf[2][DD * DD];
    const int b = blockIdx.x;
    const int lane = threadIdx.x;

    const float* P0 = Pg + (b * NCHUNK) * (DD * DD);
    for (int col = lane; col < DD; col += 32)
        #pragma unroll 4
        for (int r = 0; r < DD; ++r)
            Sbuf[0][col * DD + r] = P0[r * DD + col];
    __syncthreads();

    int cur = 0;
    #pragma unroll 1
    for (int c = 1; c < NCHUNK; ++c) {
        ssp_wave_mm64(Pg + (b * NCHUNK + c) * (DD * DD), Sbuf[cur], Sbuf[cur ^ 1]);
        __syncthreads();
        cur ^= 1;
    }

    float* out = Sout + b * (DD * DD);
    for (int col = lane; col < DD; col += 32)
        #pragma unroll 4
        for (int r = 0; r < DD; ++r)
            out[r * DD + col] = Sbuf[cur][col * DD + r];
}

// ---------------------------------------------------------------------------
// Stage D: readout.  logits = relu(S.flat @ W1 + b1) @ W2 + b2.
// grid = B, block = 64 threads (thread d computes h[d], then logit d if d<60).
// ---------------------------------------------------------------------------
__global__ void ssp_readout_kernel(const float* __restrict__ S,
                                   const float* __restrict__ W1,
                                   const float* __restrict__ b1,
                                   const float* __restrict__ W2,
                                   const float* __restrict__ b2,
                                   float* __restrict__ out) {
    __shared__ float h[DD];
    const int b = blockIdx.x;
    const int t = threadIdx.x;   // 0..63
    const float* v = S + b * DD * DD;

    float acc = b1[t];
    for (int m = 0; m < DD * DD; ++m)
        acc = fmaf(v[m], W1[m * DD + t], acc);
    h[t] = fmaxf(acc, 0.0f);
    __syncthreads();

    if (t < VV) {
        float o = b2[t];
        #pragma unroll 4
        for (int d2 = 0; d2 < DD; ++d2)
            o = fmaf(h[d2], W2[d2 * VV + t], o);
        out[b * VV + t] = o;
    }
}

// ---------------------------------------------------------------------------
extern "C" void kernel_launch(void* const* d_in, const int* in_sizes, int n_in,
                              void* d_out, int out_size, void* d_ws, size_t ws_size,
                              hipStream_t stream) {
    const int*   ids  = (const int*)d_in[0];    // [B,T]
    const float* repr = (const float*)d_in[1];  // [V,D,D]
    const float* W1   = (const float*)d_in[2];  // [D*D, D]
    const float* b1   = (const float*)d_in[3];  // [D]
    const float* W2   = (const float*)d_in[4];  // [D, V]
    const float* b2   = (const float*)d_in[5];  // [V]
    float* out = (float*)d_out;                 // [B, V]

    char* ws = (char*)d_ws;
    const size_t szQ = (size_t)VV * DD * DD * sizeof(float);             // 960 KB
    const size_t szP = (size_t)BD * NCHUNK * DD * DD * sizeof(float);    // 8 MB
    float* Qws = (float*)ws;
    float* Pws = (float*)(ws + szQ);
    float* Sws = (float*)(ws + szQ + szP);

    ssp_qr_kernel<<<VV, DD, 0, stream>>>(repr, Qws);
    ssp_chunk_kernel<<<dim3(NCHUNK, BD), 32, 0, stream>>>(ids, Qws, Pws);
    ssp_combine_kernel<<<BD, 32, 0, stream>>>(Pws, Sws);
    ssp_readout_kernel<<<BD, DD, 0, stream>>>(Sws, W1, b1, W2, b2, out);
}